// StripeAttention_55198919688397
// MI455X (gfx1250) — compile-verified
//
#include <hip/hip_runtime.h>
#include <hip/hip_bf16.h>

typedef float v2f __attribute__((ext_vector_type(2)));
typedef float v4f __attribute__((ext_vector_type(4)));
typedef float v8f __attribute__((ext_vector_type(8)));

#define B_   8
#define C_   64
#define H_   256
#define W_   256
#define LD_  8     // low dim = C/8
#define MEM_ 128   // memory bank size
#define NROWS (B_ * W_)   // 2048

// ---------------------------------------------------------------------------
// Kernel 1: per-(b,c) column mean over H.  x[b][c][h][w] -> means[(b*W+w)*C + c]
// One block per (b,c) plane (512 blocks).  256 threads = 64 w-quads x 4 h-phases.
// float4 coalesced loads; LDS combine of the 4 h-phase partials.
// ---------------------------------------------------------------------------
__global__ __launch_bounds__(256) void sa_mean_kernel(const float* __restrict__ x,
                                                      float* __restrict__ means) {
  __shared__ v4f red[256];
  const int bc = blockIdx.x;            // b*64 + c
  const int wq = threadIdx.x & 63;      // w quad index (w = wq*4)
  const int hp = threadIdx.x >> 6;      // h phase 0..3
  const float* base = x + (size_t)bc * (H_ * W_);
  v4f s = {0.f, 0.f, 0.f, 0.f};
#pragma unroll 4
  for (int h = hp; h < H_; h += 4) {
    v4f v = *(const v4f*)(base + h * W_ + wq * 4);
    s += v;
  }
  red[threadIdx.x] = s;
  __syncthreads();
  if (hp == 0) {
    v4f t = red[wq] + red[wq + 64] + red[wq + 128] + red[wq + 192];
    t *= (1.0f / (float)H_);
    const int b = bc >> 6;
    const int c = bc & 63;
    const int row = (b << 8) + wq * 4;  // b*W + w
#pragma unroll
    for (int k = 0; k < 4; ++k) means[(size_t)(row + k) * C_ + c] = t[k];
  }
}

// ---------------------------------------------------------------------------
// Kernel 2: gate MLP/attention for 16 rows per wave, fp32 WMMA 16x16x4.
//   low = Y @ w_sub + b_sub            (16x64)@(64x8)
//   f   = softmax(low @ mb * ld^-0.5)  (16x8)@(8x128)
//   y1  = f @ mb^T                     (16x128)@(128x8)
//   y2  = sigmoid(y1 @ w_up + b_up)    (16x8)@(8x64)
// Weights staged into LDS, zero-padded to N=16 where logical N is 8, so the
// B-fragment feeds are branch-free ds_loads (no exec-mask toggling).
// Fragment layouts per CDNA5 ISA 7.12.2:
//   A frag: a[i] = A[lane&15][(lane>>4)*2 + i]      (i = 0,1)
//   B frag: b[i] = B[(lane>>4)*2 + i][lane&15]
//   D frag: d[r] = D[r + (lane>>4)*8][lane&15]
// Output gates transposed to (B, C, W) layout for kernel 3.
// ---------------------------------------------------------------------------
__global__ __launch_bounds__(32) void sa_gate_kernel(const float* __restrict__ means,
                                                     const float* __restrict__ w_sub,
                                                     const float* __restrict__ b_sub,
                                                     const float* __restrict__ w_up,
                                                     const float* __restrict__ b_up,
                                                     const float* __restrict__ mb,
                                                     float* __restrict__ gT) {
  __shared__ float wsub_s[C_ * 16];    // 64x16, cols >=8 zero      (4 KB)
  __shared__ float mbT_s[MEM_ * 16];   // 128x16, cols >=8 zero     (8 KB)
  __shared__ float mb_s[LD_ * MEM_];   // 8x128 verbatim            (4 KB)
  __shared__ float wup_s[LD_ * C_];    // 8x64 verbatim             (2 KB)
  __shared__ float low_s[16 * LD_];
  __shared__ float f_s[16 * MEM_];
  __shared__ float y1_s[16 * LD_];

  const int lane = threadIdx.x;
  const int m  = lane & 15;   // A-row / B-col / D-col index for this lane
  const int hi = lane >> 4;   // K-half / M-half selector
  const int row0 = blockIdx.x * 16;

  // ---- stage weights into LDS (zero-padded where N logical = 8) ----------
  for (int idx = lane; idx < C_ * 16; idx += 32) {
    const int k = idx >> 4, n = idx & 15;
    wsub_s[idx] = (n < LD_) ? w_sub[k * LD_ + n] : 0.f;
  }
  for (int idx = lane; idx < MEM_ * 16; idx += 32) {
    const int k = idx >> 4, n = idx & 15;
    mbT_s[idx] = (n < LD_) ? mb[n * MEM_ + k] : 0.f;   // mb^T[k][n] = mb[n][k]
  }
  for (int idx = lane; idx < LD_ * MEM_; idx += 32) mb_s[idx] = mb[idx];
  for (int idx = lane; idx < LD_ * C_; idx += 32)   wup_s[idx] = w_up[idx];
  __syncthreads();

  // ---- stage 1: low = Y @ w_sub + b_sub -----------------------------------
  v8f acc1 = {};
#pragma unroll
  for (int kb = 0; kb < 16; ++kb) {
    const int k0 = kb * 4 + hi * 2;
    v2f a, b;
    a[0] = means[(size_t)(row0 + m) * C_ + k0];
    a[1] = means[(size_t)(row0 + m) * C_ + k0 + 1];
    b[0] = wsub_s[k0 * 16 + m];
    b[1] = wsub_s[(k0 + 1) * 16 + m];
    acc1 = __builtin_amdgcn_wmma_f32_16x16x4_f32(false, a, false, b,
                                                 (short)0, acc1, false, false);
  }
  if (m < LD_) {
    const float bias = b_sub[m];
#pragma unroll
    for (int r = 0; r < 8; ++r) low_s[(r + hi * 8) * LD_ + m] = acc1[r] + bias;
  }
  __syncthreads();

  // ---- stage 2: f1 = low @ mb * scale -------------------------------------
  const float scale = 0.35355339059327373f;  // 8^-0.5
#pragma unroll
  for (int j = 0; j < MEM_ / 16; ++j) {
    v8f acc2 = {};
#pragma unroll
    for (int kb = 0; kb < 2; ++kb) {
      const int k0 = kb * 4 + hi * 2;
      v2f a, b;
      a[0] = low_s[m * LD_ + k0];
      a[1] = low_s[m * LD_ + k0 + 1];
      b[0] = mb_s[k0 * MEM_ + j * 16 + m];
      b[1] = mb_s[(k0 + 1) * MEM_ + j * 16 + m];
      acc2 = __builtin_amdgcn_wmma_f32_16x16x4_f32(false, a, false, b,
                                                   (short)0, acc2, false, false);
    }
#pragma unroll
    for (int r = 0; r < 8; ++r)
      f_s[(r + hi * 8) * MEM_ + j * 16 + m] = acc2[r] * scale;
  }
  __syncthreads();

  // ---- softmax over each row of f_s (lane hi=0/1 redundantly scan row m) --
  {
    float mx = -3.402823466e38f;
    for (int i = 0; i < MEM_; ++i) mx = fmaxf(mx, f_s[m * MEM_ + i]);
    float sum = 0.f;
    for (int i = 0; i < MEM_; ++i) sum += __expf(f_s[m * MEM_ + i] - mx);
    const float inv = 1.0f / sum;
    __syncthreads();                    // all reads done before rewrite
    if (hi == 0) {
      for (int i = 0; i < MEM_; ++i)
        f_s[m * MEM_ + i] = __expf(f_s[m * MEM_ + i] - mx) * inv;
    }
  }
  __syncthreads();

  // ---- stage 3: y1 = f @ mb^T ---------------------------------------------
  v8f acc3 = {};
#pragma unroll
  for (int kb = 0; kb < 32; ++kb) {
    const int k0 = kb * 4 + hi * 2;
    v2f a, b;
    a[0] = f_s[m * MEM_ + k0];
    a[1] = f_s[m * MEM_ + k0 + 1];
    b[0] = mbT_s[k0 * 16 + m];
    b[1] = mbT_s[(k0 + 1) * 16 + m];
    acc3 = __builtin_amdgcn_wmma_f32_16x16x4_f32(false, a, false, b,
                                                 (short)0, acc3, false, false);
  }
  if (m < LD_) {
#pragma unroll
    for (int r = 0; r < 8; ++r) y1_s[(r + hi * 8) * LD_ + m] = acc3[r];
  }
  __syncthreads();

  // ---- stage 4: y2 = sigmoid(y1 @ w_up + b_up), store to (B,C,W) ----------
  const int bIdx = row0 >> 8;   // batch (row = b*W + w, W=256; tiles never straddle b)
  const int w0 = (row0 & 255) + hi * 8;
#pragma unroll
  for (int j = 0; j < C_ / 16; ++j) {
    v8f acc4 = {};
#pragma unroll
    for (int kb = 0; kb < 2; ++kb) {
      const int k0 = kb * 4 + hi * 2;
      v2f a, b;
      a[0] = y1_s[m * LD_ + k0];
      a[1] = y1_s[m * LD_ + k0 + 1];
      b[0] = wup_s[k0 * C_ + j * 16 + m];
      b[1] = wup_s[(k0 + 1) * C_ + j * 16 + m];
      acc4 = __builtin_amdgcn_wmma_f32_16x16x4_f32(false, a, false, b,
                                                   (short)0, acc4, false, false);
    }
    const int col = j * 16 + m;           // channel c
    const float bias = b_up[col];
#pragma unroll
    for (int r = 0; r < 8; ++r) {
      const float z = acc4[r] + bias;
      const float g = 1.0f / (1.0f + __expf(-z));
      gT[((size_t)(bIdx * C_ + col)) * W_ + (w0 + r)] = g;
    }
  }
}

// ---------------------------------------------------------------------------
// Kernel 3: out = x * gate * mu, streaming float4; NT stores keep x/gates in L2.
// ---------------------------------------------------------------------------
__global__ __launch_bounds__(256) void sa_apply_kernel(const float* __restrict__ x,
                                                       const float* __restrict__ gT,
                                                       const float* __restrict__ mu,
                                                       float* __restrict__ out) {
  const size_t i4 = (size_t)blockIdx.x * 256 + threadIdx.x;  // float4 index
  const size_t e = i4 * 4;                                   // element index
  const int w = (int)(e & 255);
  const int bch = (int)(e >> 8);   // bc*256 + h
  const int bc = bch >> 8;         // b*64 + c
  v4f xv = *(const v4f*)(x + e);
  v4f gv = *(const v4f*)(gT + ((size_t)bc << 8) + w);
  const float muv = mu[0];
  v4f o = xv * gv * muv;
  __builtin_nontemporal_store(o, (v4f*)(out + e));
}

// ---------------------------------------------------------------------------
extern "C" void kernel_launch(void* const* d_in, const int* in_sizes, int n_in,
                              void* d_out, int out_size, void* d_ws, size_t ws_size,
                              hipStream_t stream) {
  const float* x     = (const float*)d_in[0];
  const float* w_sub = (const float*)d_in[1];
  const float* b_sub = (const float*)d_in[2];
  const float* w_up  = (const float*)d_in[3];
  const float* b_up  = (const float*)d_in[4];
  const float* mb    = (const float*)d_in[5];
  const float* mu    = (const float*)d_in[6];
  float* out = (float*)d_out;

  float* means = (float*)d_ws;                 // (2048, 64)   512 KB
  float* gT    = means + (size_t)NROWS * C_;   // (B*C, W)     512 KB

  sa_mean_kernel<<<B_ * C_, 256, 0, stream>>>(x, means);
  sa_gate_kernel<<<NROWS / 16, 32, 0, stream>>>(means, w_sub, b_sub, w_up, b_up, mb, gT);
  const int total4 = (B_ * C_ * H_ * W_) / 4;  // 8,388,608
  sa_apply_kernel<<<total4 / 256, 256, 0, stream>>>(x, gT, mu, out);
}